// Net_13718125543828
// MI455X (gfx1250) — compile-verified
//
#include <hip/hip_runtime.h>
#include <hip/hip_bf16.h>
#include <math.h>

// ---------------- sizes (match reference) ----------------
static constexpr int N1 = 10000, E1 = 60000;
static constexpr int N2 = 2500,  E2 = 15000;
static constexpr int N3 = 625,   E3 = 3750;
static constexpr int N4 = 160,   E4 = 960;
static constexpr int NC = 4;
static constexpr int KK = 125;   // 5^3 spline kernels

typedef float v2f  __attribute__((ext_vector_type(2)));
typedef float v8f  __attribute__((ext_vector_type(8)));

// ---------------- utility kernels ----------------
__global__ void k_zero(float* p, int n) {
  int t = blockIdx.x * blockDim.x + threadIdx.x;
  if (t < n) p[t] = 0.0f;
}
__global__ void k_zero_i(int* p, int n) {
  int t = blockIdx.x * blockDim.x + threadIdx.x;
  if (t < n) p[t] = 0;
}

// ---------------- spline corner helper ----------------
__device__ __forceinline__ void spline_corners(const float* __restrict__ u, int e,
                                               int* idx, float* wgt) {
  float v0 = u[e * 3 + 0] * 4.0f;
  float v1 = u[e * 3 + 1] * 4.0f;
  float v2 = u[e * 3 + 2] * 4.0f;
  float k0f = fminf(fmaxf(floorf(v0), 0.0f), 3.0f);
  float k1f = fminf(fmaxf(floorf(v1), 0.0f), 3.0f);
  float k2f = fminf(fmaxf(floorf(v2), 0.0f), 3.0f);
  float f0 = v0 - k0f, f1 = v1 - k1f, f2 = v2 - k2f;
  int k0 = (int)k0f, k1 = (int)k1f, k2 = (int)k2f;
#pragma unroll
  for (int c = 0; c < 8; ++c) {
    int b0 = c & 1, b1 = (c >> 1) & 1, b2 = (c >> 2) & 1;
    idx[c] = (k0 + b0) * 25 + (k1 + b1) * 5 + (k2 + b2);
    wgt[c] = (b0 ? f0 : 1.0f - f0) * (b1 ? f1 : 1.0f - f1) * (b2 ? f2 : 1.0f - f2);
  }
}

// ---------------- bucketing: histogram (LDS two-level) ----------------
__global__ void k_pair_count(const float* __restrict__ u, int E, int* __restrict__ counts) {
  __shared__ int sc[KK];
  const int t = threadIdx.x;
  for (int k = t; k < KK; k += blockDim.x) sc[k] = 0;
  __syncthreads();
  const int e = blockIdx.x * blockDim.x + t;
  if (e < E) {
    int idx[8]; float wgt[8];
    spline_corners(u, e, idx, wgt);
#pragma unroll
    for (int c = 0; c < 8; ++c) atomicAdd(&sc[idx[c]], 1);
  }
  __syncthreads();
  for (int k = t; k < KK; k += blockDim.x)
    if (sc[k]) atomicAdd(&counts[k], sc[k]);
}

// padded exclusive scan over 125 buckets (trivial size -> serial)
// buckets padded to 64 rows so one block owns 4 row tiles of a single bucket
__global__ void k_scan_offsets(const int* __restrict__ counts, int* __restrict__ offs,
                               int* __restrict__ cursor) {
  if (blockIdx.x == 0 && threadIdx.x == 0) {
    int off = 0;
    for (int k = 0; k < KK; ++k) {
      offs[k] = off;
      cursor[k] = off;
      off += (counts[k] + 63) & ~63;
    }
    offs[KK] = off;
  }
}

__global__ void k_pair_init(int* __restrict__ pair_edge, float* __restrict__ pair_wgt, int cap) {
  int t = blockIdx.x * blockDim.x + threadIdx.x;
  if (t < cap) { pair_edge[t] = -1; pair_wgt[t] = 0.0f; }
}

// two-level scatter: LDS local ranks + one global reservation per (bucket, block)
__global__ void k_pair_scatter(const float* __restrict__ u, int E, int* __restrict__ cursor,
                               int* __restrict__ pair_edge, float* __restrict__ pair_wgt) {
  __shared__ int sc[KK];
  __shared__ int sbase[KK];
  const int t = threadIdx.x;
  for (int k = t; k < KK; k += blockDim.x) sc[k] = 0;
  __syncthreads();
  const int e = blockIdx.x * blockDim.x + t;
  int idx[8]; float wgt[8]; int lrank[8];
  const bool valid = e < E;
  if (valid) {
    spline_corners(u, e, idx, wgt);
#pragma unroll
    for (int c = 0; c < 8; ++c) lrank[c] = atomicAdd(&sc[idx[c]], 1);
  }
  __syncthreads();
  for (int k = t; k < KK; k += blockDim.x)
    sbase[k] = sc[k] ? atomicAdd(&cursor[k], sc[k]) : 0;
  __syncthreads();
  if (valid) {
#pragma unroll
    for (int c = 0; c < 8; ++c) {
      int pos = sbase[idx[c]] + lrank[c];
      pair_edge[pos] = e;
      pair_wgt[pos] = wgt[c];
    }
  }
}

__global__ void k_deg(const int* __restrict__ ei, float* __restrict__ deg, int E) {
  int e = blockIdx.x * blockDim.x + threadIdx.x;
  if (e < E) atomicAdd(&deg[ei[E + e]], 1.0f);
}

// ---------------- bucketed message GEMM on WMMA --------------------------------------
// One block = OUT/16 waves, owns 64 pairs (4 row tiles) of one bucket.
// A-panel (wgt * x[src], 64 x IN) staged to LDS once; each wave loads its B fragment
// once per K-step and reuses it across 4 independent accumulator chains.
template <int IN, int OUT>
__global__ void k_bucket_gemm_t(const int* __restrict__ offs,
                                const int* __restrict__ pair_edge,
                                const float* __restrict__ pair_wgt,
                                const int* __restrict__ ei,
                                const float* __restrict__ x,
                                const float* __restrict__ W,
                                float* __restrict__ agg, int E) {
  constexpr int NW = OUT / 16;               // waves per block
  constexpr int RT = 4;                      // row tiles per block
  __shared__ float As[64 * IN];
  __shared__ float swg[64];
  __shared__ int   ssrc[64];
  __shared__ int   sdst[64];

  const int base = blockIdx.x * 64;
  const int total = offs[KK];
  if (base >= total) return;                 // block-uniform exit
  // binary search: offs[kk] <= base < offs[kk+1]
  int lo = 0, hi = KK;
  while (hi - lo > 1) { int mid = (lo + hi) >> 1; if (offs[mid] <= base) lo = mid; else hi = mid; }
  const int kk = lo;

  const int t = threadIdx.x;
  for (int i = t; i < 64; i += NW * 32) {
    const int e = pair_edge[base + i];
    sdst[i] = (e >= 0) ? ei[E + e] : -1;
    ssrc[i] = (e >= 0) ? ei[e] : 0;
    swg[i]  = (e >= 0) ? pair_wgt[base + i] : 0.0f;
  }
  __syncthreads();
  for (int i = t; i < 64 * IN; i += NW * 32) {
    const int r = i / IN, k = i % IN;        // IN is compile-time
    As[i] = swg[r] * x[(size_t)ssrc[r] * IN + k];
  }
  __syncthreads();

  const int lane = t & 31;
  const int row = lane & 15;
  const int hi16 = lane >> 4;
  const int kb = hi16 << 1;                  // K sub-offset 0 or 2
  const int tn = (t >> 5) * 16;              // this wave's column tile
  const float* __restrict__ Wk = W + (size_t)kk * IN * OUT + tn;

  v8f acc[RT] = {};
#pragma unroll
  for (int k0 = 0; k0 < IN; k0 += 4) {
    const int k = k0 + kb;
    v2f b;
    if (k0 + 4 <= IN) {                      // compile-time after unroll
      b.x = Wk[(size_t)k * OUT + row];
      b.y = Wk[(size_t)(k + 1) * OUT + row];
    } else {                                 // tail only for IN % 4 != 0
      b.x = (k     < IN) ? Wk[(size_t)k       * OUT + row] : 0.0f;
      b.y = (k + 1 < IN) ? Wk[(size_t)(k + 1) * OUT + row] : 0.0f;
    }
#pragma unroll
    for (int r = 0; r < RT; ++r) {
      v2f a;
      const int arow = (r * 16 + row) * IN;
      if (k0 + 4 <= IN) {
        a.x = As[arow + k];
        a.y = As[arow + k + 1];
      } else {
        a.x = (k     < IN) ? As[arow + k]     : 0.0f;
        a.y = (k + 1 < IN) ? As[arow + k + 1] : 0.0f;
      }
      acc[r] = __builtin_amdgcn_wmma_f32_16x16x4_f32(false, a, false, b, (short)0,
                                                     acc[r], false, false);
    }
  }

  const int col = tn + row;
#pragma unroll
  for (int r = 0; r < RT; ++r) {
#pragma unroll
    for (int j = 0; j < 8; ++j) {
      const int dst = sdst[r * 16 + j + (hi16 << 3)];
      if (dst >= 0) atomicAdd(&agg[(size_t)dst * OUT + col], acc[r][j]);
    }
  }
}

// ---------------- dense WMMA tiled GEMM: C[M,N] = A[M,K] @ B[K,N], row-major --------
template <int K>
__global__ void k_gemm_wmma_t(const float* __restrict__ A, const float* __restrict__ B,
                              float* __restrict__ C, int M, int N) {
  const int tm = blockIdx.x * 16;
  const int tn = blockIdx.y * 16;
  const int lane = threadIdx.x & 31;
  const int row = lane & 15;
  const int hi = lane >> 4;
  const int kb = hi << 1;
  v8f acc = {};
  const int ar = tm + row;
  const int bc = tn + row;
  if ((tm + 16 <= M) && (tn + 16 <= N)) {
    // interior tile: fully unguarded, unrolled
#pragma unroll
    for (int k0 = 0; k0 < K; k0 += 4) {
      v2f a, b;
      const int k = k0 + kb;
      if (k0 + 4 <= K) {
        a.x = A[(size_t)ar * K + k];
        a.y = A[(size_t)ar * K + k + 1];
        b.x = B[(size_t)k * N + bc];
        b.y = B[(size_t)(k + 1) * N + bc];
      } else {
        a.x = (k     < K) ? A[(size_t)ar * K + k]     : 0.0f;
        a.y = (k + 1 < K) ? A[(size_t)ar * K + k + 1] : 0.0f;
        b.x = (k     < K) ? B[(size_t)k * N + bc]       : 0.0f;
        b.y = (k + 1 < K) ? B[(size_t)(k + 1) * N + bc] : 0.0f;
      }
      acc = __builtin_amdgcn_wmma_f32_16x16x4_f32(false, a, false, b, (short)0, acc,
                                                  false, false);
    }
#pragma unroll
    for (int j = 0; j < 8; ++j)
      C[(size_t)(tm + j + (hi << 3)) * N + tn + row] = acc[j];
  } else {
    // boundary tile: guarded
#pragma unroll
    for (int k0 = 0; k0 < K; k0 += 4) {
      v2f a = {0.0f, 0.0f}, b = {0.0f, 0.0f};
      const int k = k0 + kb;
      if (ar < M) {
        if (k < K)     a.x = A[(size_t)ar * K + k];
        if (k + 1 < K) a.y = A[(size_t)ar * K + k + 1];
      }
      if (bc < N) {
        if (k < K)     b.x = B[(size_t)k * N + bc];
        if (k + 1 < K) b.y = B[(size_t)(k + 1) * N + bc];
      }
      acc = __builtin_amdgcn_wmma_f32_16x16x4_f32(false, a, false, b, (short)0, acc,
                                                  false, false);
    }
    if (bc < N) {
#pragma unroll
      for (int j = 0; j < 8; ++j) {
        const int r = tm + j + (hi << 3);
        if (r < M) C[(size_t)r * N + bc] = acc[j];
      }
    }
  }
}

// out = elu(agg/max(deg,1) + root + bias)
__global__ void k_finalize(const float* __restrict__ agg, const float* __restrict__ deg,
                           const float* __restrict__ root, const float* __restrict__ bias,
                           float* __restrict__ out, int N, int OUT) {
  int t = blockIdx.x * blockDim.x + threadIdx.x;
  if (t >= N * OUT) return;
  int n = t / OUT, o = t % OUT;
  float d = fmaxf(deg[n], 1.0f);
  float v = agg[t] / d + root[t] + bias[o];
  out[t] = v > 0.0f ? v : (expf(v) - 1.0f);
}

// ---------------- segment-max pooling (order-preserving uint encoding) ----------------
__device__ __forceinline__ unsigned enc_f(float f) {
  unsigned b = __float_as_uint(f);
  return (b & 0x80000000u) ? ~b : (b | 0x80000000u);
}
__device__ __forceinline__ float dec_f(unsigned c) {
  return (c & 0x80000000u) ? __uint_as_float(c ^ 0x80000000u) : __uint_as_float(~c);
}
__global__ void k_pool_init(unsigned* p, int n) {
  int t = blockIdx.x * blockDim.x + threadIdx.x;
  if (t < n) p[t] = 0x00800000u;  // encoding of -FLT_MAX
}
__global__ void k_pool_max(const float* __restrict__ h, const int* __restrict__ clus,
                           unsigned* __restrict__ pool, int Nin, int C) {
  int t = blockIdx.x * blockDim.x + threadIdx.x;
  if (t >= Nin * C) return;
  int n = t / C, c = t % C;
  atomicMax(&pool[(size_t)clus[n] * C + c], enc_f(h[t]));
}
__global__ void k_pool_decode_cat(const unsigned* __restrict__ pool, float* __restrict__ out,
                                  int Nn, int C) {
  int t = blockIdx.x * blockDim.x + threadIdx.x;
  if (t >= Nn * (C + 1)) return;
  int n = t / (C + 1), j = t % (C + 1);
  out[t] = (j == C) ? 1.0f : dec_f(pool[(size_t)n * C + j]);
}

__global__ void k_gather_add_bias(const float* __restrict__ up, const int* __restrict__ clus,
                                  const float* __restrict__ skt, const float* __restrict__ skb,
                                  float* __restrict__ out, int N, int C) {
  int t = blockIdx.x * blockDim.x + threadIdx.x;
  if (t >= N * C) return;
  int n = t / C, c = t % C;
  out[t] = up[(size_t)clus[n] * C + c] + skt[t] + skb[c];
}

__global__ void k_bias_elu(const float* __restrict__ in, const float* __restrict__ b,
                           float* __restrict__ out, int N, int C) {
  int t = blockIdx.x * blockDim.x + threadIdx.x;
  if (t >= N * C) return;
  float v = in[t] + b[t % C];
  out[t] = v > 0.0f ? v : (expf(v) - 1.0f);
}

__global__ void k_logsoftmax(const float* __restrict__ logits, const float* __restrict__ b,
                             float* __restrict__ out, int N, int C) {
  int n = blockIdx.x * blockDim.x + threadIdx.x;
  if (n >= N) return;
  float v[NC];
  float m = -1e30f;
  for (int c = 0; c < C; ++c) { v[c] = logits[n * C + c] + b[c]; m = fmaxf(m, v[c]); }
  float s = 0.0f;
  for (int c = 0; c < C; ++c) s += expf(v[c] - m);
  float l = logf(s);
  for (int c = 0; c < C; ++c) out[n * C + c] = v[c] - m - l;
}

// ---------------- host orchestration ----------------
extern "C" void kernel_launch(void* const* d_in, const int* in_sizes, int n_in,
                              void* d_out, int out_size, void* d_ws, size_t ws_size,
                              hipStream_t stream) {
  (void)in_sizes; (void)n_in; (void)out_size; (void)ws_size;
  const float* x  = (const float*)d_in[0];
  const int* ei1  = (const int*)d_in[1];  const float* u1 = (const float*)d_in[2];
  const int* ei2  = (const int*)d_in[3];  const float* u2 = (const float*)d_in[4];
  const int* ei3  = (const int*)d_in[5];  const float* u3 = (const float*)d_in[6];
  const int* ei4  = (const int*)d_in[7];  const float* u4 = (const float*)d_in[8];
  const int* c1   = (const int*)d_in[9];
  const int* c2   = (const int*)d_in[10];
  const int* c3   = (const int*)d_in[11];
  const float *w1  = (const float*)d_in[12], *r1  = (const float*)d_in[13], *b1  = (const float*)d_in[14];
  const float *w12 = (const float*)d_in[15], *r12 = (const float*)d_in[16], *b12 = (const float*)d_in[17];
  const float *w2  = (const float*)d_in[18], *r2  = (const float*)d_in[19], *b2  = (const float*)d_in[20];
  const float *w22 = (const float*)d_in[21], *r22 = (const float*)d_in[22], *b22 = (const float*)d_in[23];
  const float *w3  = (const float*)d_in[24], *r3  = (const float*)d_in[25], *b3  = (const float*)d_in[26];
  const float *w32 = (const float*)d_in[27], *r32 = (const float*)d_in[28], *b32 = (const float*)d_in[29];
  const float *w4  = (const float*)d_in[30], *r4  = (const float*)d_in[31], *b4  = (const float*)d_in[32];
  const float *w42 = (const float*)d_in[33], *r42 = (const float*)d_in[34], *b42 = (const float*)d_in[35];
  const float *w5  = (const float*)d_in[36], *r5  = (const float*)d_in[37], *b5  = (const float*)d_in[38];
  const float *w6  = (const float*)d_in[39], *r6  = (const float*)d_in[40], *b6  = (const float*)d_in[41];
  const float *w7  = (const float*)d_in[42], *r7  = (const float*)d_in[43], *b7  = (const float*)d_in[44];
  const float *fc1w = (const float*)d_in[45], *fc1b = (const float*)d_in[46];
  const float *fc2w = (const float*)d_in[47], *fc2b = (const float*)d_in[48];
  const float *sk1w = (const float*)d_in[49], *sk1b = (const float*)d_in[50];
  const float *sk2w = (const float*)d_in[51], *sk2b = (const float*)d_in[52];
  const float *sk3w = (const float*)d_in[53], *sk3b = (const float*)d_in[54];

  // bump allocator over workspace (floats)
  float* base = (float*)d_ws;
  size_t off = 0;
  auto alloc = [&](size_t n) { float* p = base + off; off += n; return p; };

  float* h1a   = alloc((size_t)N1 * 64);
  float* h1    = alloc((size_t)N1 * 64);
  float* tbuf  = alloc((size_t)N1 * 64);   // root / gemm scratch
  float* agg   = alloc((size_t)N1 * 64);
  float* deg   = alloc((size_t)N1);
  unsigned* poolu = (unsigned*)alloc((size_t)N2 * 64);
  float* h2cat = alloc((size_t)N2 * 65);
  float* h2a   = alloc((size_t)N2 * 64);
  float* h2    = alloc((size_t)N2 * 64);
  float* h3cat = alloc((size_t)N3 * 65);
  float* h3a   = alloc((size_t)N3 * 64);
  float* h3    = alloc((size_t)N3 * 64);
  float* h4cat = alloc((size_t)N4 * 65);
  float* h4a   = alloc((size_t)N4 * 64);
  float* h4b   = alloc((size_t)N4 * 64);
  float* h4    = alloc((size_t)N4 * 32);
  float* g3in  = alloc((size_t)N3 * 32);
  float* g3    = alloc((size_t)N3 * 32);
  float* g2in  = alloc((size_t)N2 * 32);
  float* g2    = alloc((size_t)N2 * 32);
  float* g1in  = alloc((size_t)N1 * 32);
  float* g1    = alloc((size_t)N1 * 32);
  float* logit = alloc((size_t)N1 * NC);
  float* skt   = alloc((size_t)N1 * 32);   // skip/fc gemm scratch
  // bucketing scratch (sized for level 1, 64-padded buckets)
  const int pairCap = 8 * E1 + 64 * KK;
  int*   counts    = (int*)alloc(KK);
  int*   offs      = (int*)alloc(KK + 1);
  int*   cursor    = (int*)alloc(KK);
  int*   pair_edge = (int*)alloc((size_t)pairCap);
  float* pair_wgt  = alloc((size_t)pairCap);

  auto gemm = [&](const float* A, const float* B, float* C, int M, int N, int K) {
    dim3 g((M + 15) / 16, (N + 15) / 16);
    switch (K) {
      case 1:  k_gemm_wmma_t<1><<<g, dim3(32), 0, stream>>>(A, B, C, M, N);  break;
      case 32: k_gemm_wmma_t<32><<<g, dim3(32), 0, stream>>>(A, B, C, M, N); break;
      case 64: k_gemm_wmma_t<64><<<g, dim3(32), 0, stream>>>(A, B, C, M, N); break;
      case 65: k_gemm_wmma_t<65><<<g, dim3(32), 0, stream>>>(A, B, C, M, N); break;
      default: break;
    }
  };

  auto conv = [&](const float* xin, const int* ei, const float* u, const float* W,
                  const float* R, const float* bb, float* out, int N, int E, int IN, int OUT) {
    const int nTot = N * OUT;
    k_zero<<<(nTot + 255) / 256, 256, 0, stream>>>(agg, nTot);
    k_zero<<<(N + 255) / 256, 256, 0, stream>>>(deg, N);
    k_zero_i<<<1, 256, 0, stream>>>(counts, KK);
    gemm(xin, R, tbuf, N, OUT, IN);  // root transform on WMMA
    // bucket (edge, corner) pairs by spline kernel index
    k_pair_count<<<(E + 255) / 256, 256, 0, stream>>>(u, E, counts);
    k_scan_offsets<<<1, 1, 0, stream>>>(counts, offs, cursor);
    const int cap = 8 * E + 64 * KK;
    k_pair_init<<<(cap + 255) / 256, 256, 0, stream>>>(pair_edge, pair_wgt, cap);
    k_pair_scatter<<<(E + 255) / 256, 256, 0, stream>>>(u, E, cursor, pair_edge, pair_wgt);
    k_deg<<<(E + 255) / 256, 256, 0, stream>>>(ei, deg, E);
    // per-bucket WMMA GEMM: one block per 64-pair group (device-side exit past total)
    const int maxBlocks = (8 * E) / 64 + KK;
    dim3 g(maxBlocks);
    if (IN == 1 && OUT == 64)
      k_bucket_gemm_t<1, 64><<<g, dim3(128), 0, stream>>>(offs, pair_edge, pair_wgt, ei, xin, W, agg, E);
    else if (IN == 64 && OUT == 64)
      k_bucket_gemm_t<64, 64><<<g, dim3(128), 0, stream>>>(offs, pair_edge, pair_wgt, ei, xin, W, agg, E);
    else if (IN == 65 && OUT == 64)
      k_bucket_gemm_t<65, 64><<<g, dim3(128), 0, stream>>>(offs, pair_edge, pair_wgt, ei, xin, W, agg, E);
    else if (IN == 32 && OUT == 32)
      k_bucket_gemm_t<32, 32><<<g, dim3(64), 0, stream>>>(offs, pair_edge, pair_wgt, ei, xin, W, agg, E);
    k_finalize<<<(nTot + 255) / 256, 256, 0, stream>>>(agg, deg, tbuf, bb, out, N, OUT);
  };

  auto pool = [&](const float* h, const int* clus, float* outcat, int Nin, int Nout, int C) {
    k_pool_init<<<(Nout * C + 255) / 256, 256, 0, stream>>>(poolu, Nout * C);
    k_pool_max<<<(Nin * C + 255) / 256, 256, 0, stream>>>(h, clus, poolu, Nin, C);
    k_pool_decode_cat<<<(Nout * (C + 1) + 255) / 256, 256, 0, stream>>>(poolu, outcat, Nout, C);
  };

  auto skip = [&](const float* up, const int* clus, const float* h, const float* skw,
                  const float* skb, float* out, int N, int CIN, int COUT) {
    gemm(h, skw, skt, N, COUT, CIN);
    k_gather_add_bias<<<(N * COUT + 255) / 256, 256, 0, stream>>>(up, clus, skt, skb, out, N, COUT);
  };

  // -------- encoder --------
  conv(x,    ei1, u1, w1,  r1,  b1,  h1a, N1, E1, 1,  64);
  conv(h1a,  ei1, u1, w12, r12, b12, h1,  N1, E1, 64, 64);
  pool(h1, c1, h2cat, N1, N2, 64);
  conv(h2cat, ei2, u2, w2,  r2,  b2,  h2a, N2, E2, 65, 64);
  conv(h2a,   ei2, u2, w22, r22, b22, h2,  N2, E2, 64, 64);
  pool(h2, c2, h3cat, N2, N3, 64);
  conv(h3cat, ei3, u3, w3,  r3,  b3,  h3a, N3, E3, 65, 64);
  conv(h3a,   ei3, u3, w32, r32, b32, h3,  N3, E3, 64, 64);
  pool(h3, c3, h4cat, N3, N4, 64);
  conv(h4cat, ei4, u4, w4,  r4,  b4,  h4a, N4, E4, 65, 64);
  conv(h4a,   ei4, u4, w42, r42, b42, h4b, N4, E4, 64, 64);
  conv(h4b,   ei4, u4, w42, r42, b42, h4a, N4, E4, 64, 64);  // conv42 applied twice

  // fc1 + elu
  gemm(h4a, fc1w, skt, N4, 32, 64);
  k_bias_elu<<<(N4 * 32 + 255) / 256, 256, 0, stream>>>(skt, fc1b, h4, N4, 32);

  // -------- decoder --------
  skip(h4, c3, h3, sk3w, sk3b, g3in, N3, 64, 32);
  conv(g3in, ei3, u3, w5, r5, b5, g3, N3, E3, 32, 32);
  skip(g3, c2, h2, sk2w, sk2b, g2in, N2, 64, 32);
  conv(g2in, ei2, u2, w6, r6, b6, g2, N2, E2, 32, 32);
  skip(g2, c1, h1, sk1w, sk1b, g1in, N1, 64, 32);
  conv(g1in, ei1, u1, w7, r7, b7, g1, N1, E1, 32, 32);

  // classifier + log_softmax
  gemm(g1, fc2w, logit, N1, NC, 32);
  k_logsoftmax<<<(N1 + 255) / 256, 256, 0, stream>>>(logit, fc2b, (float*)d_out, N1, NC);
}